// HybridParallelClassifier_61959198212623
// MI455X (gfx1250) — compile-verified
//
#include <hip/hip_runtime.h>
#include <math.h>

// ---- model dims (match reference) ----
#define B_      256
#define L_      1000
#define C_IN    12
#define D_MODEL 128
#define D_STATE 16
#define D_CONV  4
#define D_INNER 256          // EXPAND * D_MODEL
#define DT_RANK 8
#define H_LSTM  128
#define FC_H    128
#define NCLS    5
#define G4      (4 * H_LSTM) // 512 gate width

typedef __attribute__((ext_vector_type(2))) float v2f;
typedef __attribute__((ext_vector_type(8))) float v8f;

// CDNA5 fp32 tensor op: D(16x16) = A(16x4) * B(4x16) + C. Wave32; A/B = 2 VGPRs/lane.
__device__ __forceinline__ v8f wmma4(v2f a, v2f b, v8f c) {
  return __builtin_amdgcn_wmma_f32_16x16x4_f32(false, a, false, b, (short)0, c, false, false);
}

__device__ __forceinline__ float sigmoidf_(float x) { return 1.0f / (1.0f + __expf(-x)); }
__device__ __forceinline__ float siluf_(float x)    { return x * sigmoidf_(x); }

// ---------------------------------------------------------------------------
// Kernel 0: fold mp into in_W  (Wc = in_W @ mp_W : [512,12], bc = in_W @ mp_b)
// and fold LSTM biases (bsum = b_ih + b_hh).
// ---------------------------------------------------------------------------
__global__ __launch_bounds__(512) void k_fold(
    const float* __restrict__ mp_W, const float* __restrict__ mp_b,
    const float* __restrict__ in_W,
    const float* __restrict__ b_ih, const float* __restrict__ b_hh,
    float* __restrict__ Wc, float* __restrict__ bc, float* __restrict__ bsum)
{
  const int j = threadIdx.x;                // 0..511 (row of in_W / W_ih space)
  float acc[C_IN];
#pragma unroll
  for (int i = 0; i < C_IN; ++i) acc[i] = 0.0f;
  float accb = 0.0f;
  for (int k = 0; k < D_MODEL; ++k) {
    const float w = in_W[j * D_MODEL + k];
    accb += w * mp_b[k];
#pragma unroll
    for (int i = 0; i < C_IN; ++i) acc[i] += w * mp_W[k * C_IN + i];
  }
#pragma unroll
  for (int i = 0; i < C_IN; ++i) Wc[j * C_IN + i] = acc[i];
  bc[j]   = accb;
  bsum[j] = b_ih[j] + b_hh[j];
}

// ---------------------------------------------------------------------------
// Kernel 1: fully fused mamba branch. 1 WG per batch element, 256 threads =
// 256 inner channels. Per t: folded in-proj (K=12), rolling causal conv,
// SiLU, xp projection (tree-reduced in LDS), dt softplus, selective scan
// (h[16] in registers), SiLU(z) gate, running mean accumulation.
// ---------------------------------------------------------------------------
__global__ __launch_bounds__(256) void k_mamba(
    const float* __restrict__ x,
    const float* __restrict__ Wc, const float* __restrict__ bc,
    const float* __restrict__ conv_w, const float* __restrict__ conv_b,
    const float* __restrict__ xp_W, const float* __restrict__ dt_W,
    const float* __restrict__ dt_b, const float* __restrict__ A_log,
    const float* __restrict__ Dvec, float* __restrict__ ysum_ws)
{
  const int b = blockIdx.x;
  const int d = threadIdx.x;                // inner channel 0..255

  __shared__ float xs[C_IN];                // x row for current t
  __shared__ float us[D_INNER];             // SiLU(conv) activations
  __shared__ float xpart[40 * 4];           // partial sums for xp projection
  __shared__ float xdbc_s[40];              // [dt_in(8) | B(16) | C(16)]

  // Per-thread constant loads (held in VGPRs across the whole time loop).
  float wcu[C_IN], wcz[C_IN];
#pragma unroll
  for (int i = 0; i < C_IN; ++i) {
    wcu[i] = Wc[d * C_IN + i];
    wcz[i] = Wc[(D_INNER + d) * C_IN + i];
  }
  const float bcu = bc[d], bcz = bc[D_INNER + d];
  const float cw0 = conv_w[d * D_CONV + 0], cw1 = conv_w[d * D_CONV + 1];
  const float cw2 = conv_w[d * D_CONV + 2], cw3 = conv_w[d * D_CONV + 3];
  const float cb  = conv_b[d];
  float dtw[DT_RANK];
#pragma unroll
  for (int r = 0; r < DT_RANK; ++r) dtw[r] = dt_W[d * DT_RANK + r];
  const float dtb = dt_b[d];
  float Areg[D_STATE];
#pragma unroll
  for (int n = 0; n < D_STATE; ++n) Areg[n] = -__expf(A_log[d * D_STATE + n]);
  const float Dd = Dvec[d];

  float h[D_STATE];
#pragma unroll
  for (int n = 0; n < D_STATE; ++n) h[n] = 0.0f;
  float ysum = 0.0f;
  float ur0 = 0.f, ur1 = 0.f, ur2 = 0.f, ur3 = 0.f;  // rolling conv window

  const float* xb = x + (size_t)b * L_ * C_IN;

  for (int t = 0; t < L_; ++t) {
    if (d < C_IN) xs[d] = xb[t * C_IN + d];
    __syncthreads();

    // folded in-projection (K=12) -> u_raw, z_raw
    float uraw = bcu, zraw = bcz;
#pragma unroll
    for (int i = 0; i < C_IN; ++i) {
      uraw += xs[i] * wcu[i];
      zraw += xs[i] * wcz[i];
    }
    // causal depthwise conv (window 4), then SiLU
    ur0 = ur1; ur1 = ur2; ur2 = ur3; ur3 = uraw;
    const float uc = cb + ur0 * cw0 + ur1 * cw1 + ur2 * cw2 + ur3 * cw3;
    const float ua = siluf_(uc);
    us[d] = ua;
    // prefetch next x row while we are at it (global_prefetch_b8)
    if (d < C_IN && t + 1 < L_) __builtin_prefetch(&xb[(t + 1) * C_IN + d], 0, 0);
    __syncthreads();

    // xp projection: 40 outputs x K=256, split into 4 partials of 64
    if (d < 160) {
      const int j = d >> 2, p = d & 3;
      const float* w  = xp_W + j * D_INNER + p * 64;
      const float* uu = us + p * 64;
      float s = 0.0f;
#pragma unroll 8
      for (int q = 0; q < 64; ++q) s += uu[q] * w[q];
      xpart[d] = s;
    }
    __syncthreads();
    if (d < 40)
      xdbc_s[d] = xpart[d * 4 + 0] + xpart[d * 4 + 1] + xpart[d * 4 + 2] + xpart[d * 4 + 3];
    __syncthreads();

    // dt = softplus(dt_in @ dt_Wᵀ + dt_b)
    float dl = dtb;
#pragma unroll
    for (int r = 0; r < DT_RANK; ++r) dl += xdbc_s[r] * dtw[r];
    const float dt  = (dl > 20.0f) ? dl : __logf(1.0f + __expf(dl));
    const float dtu = dt * ua;

    // selective scan step + output contraction with C
    float y = 0.0f;
#pragma unroll
    for (int n = 0; n < D_STATE; ++n) {
      const float dA = __expf(dt * Areg[n]);
      h[n] = dA * h[n] + dtu * xdbc_s[DT_RANK + n];
      y += h[n] * xdbc_s[DT_RANK + D_STATE + n];
    }
    y += ua * Dd;          // skip connection
    y *= siluf_(zraw);     // gate
    ysum += y;             // mean-pool accumulator (divide by L in head)
  }
  ysum_ws[b * D_INNER + d] = ysum;
}

// ---------------------------------------------------------------------------
// Kernel 2: LSTM branch on fp32 WMMA. 16 batch rows / WG, 32 waves; wave w
// owns gate columns [16w,16w+16) with its W_hh slice resident in VGPRs
// (32 x v2f). Per step: 3 WMMA k-steps for x@W_ihᵀ (K=12) + 32 WMMA k-steps
// for h@W_hhᵀ (K=128), gates in LDS, cell state in registers.
// ---------------------------------------------------------------------------
#define HSTR 132   // 132 % 64 == 4 -> conflict-free column reads of h
#define GSTR 516   // 516 % 64 == 4 -> conflict-free g tile stores/reads

__global__ __launch_bounds__(1024) void k_lstm(
    const float* __restrict__ x,
    const float* __restrict__ W_ih, const float* __restrict__ W_hh,
    const float* __restrict__ bsum, float* __restrict__ hsum_ws)
{
  const int tid  = threadIdx.x;
  const int b0   = blockIdx.x * 16;       // batch tile base
  const int lane = tid & 31;
  const int wv   = tid >> 5;              // wave id 0..31 = N tile
  const int col  = lane & 15;
  const int kb   = (lane >> 4) * 2;       // 0 | 2 (fp32 16x4 A/B layout)
  const int n    = wv * 16 + col;         // global gate column 0..511
  const int arow = lane & 15;             // A-matrix row = batch row in tile
  const int rhi  = (lane >> 4) * 8;       // C/D row offset

  __shared__ float xs[16][C_IN];
  __shared__ float h_lds[16][HSTR];
  __shared__ float g_lds[16][GSTR];

  // Resident B-tiles: W_hhᵀ slice for this wave's 16 columns (64 VGPRs/lane)
  v2f bh[32];
#pragma unroll
  for (int kk = 0; kk < 32; ++kk) {
    bh[kk].x = W_hh[n * H_LSTM + 4 * kk + kb];
    bh[kk].y = W_hh[n * H_LSTM + 4 * kk + kb + 1];
  }
  // W_ihᵀ slice: K=12 = exactly 3 k-steps of 4
  v2f bxw[3];
#pragma unroll
  for (int kk = 0; kk < 3; ++kk) {
    bxw[kk].x = W_ih[n * C_IN + 4 * kk + kb];
    bxw[kk].y = W_ih[n * C_IN + 4 * kk + kb + 1];
  }
  const float biasn = bsum[n];

  // Gate-phase cell ownership: 2048 cells / 1024 threads = 2 cells/thread
  const int cb_ = tid >> 7;               // batch row 0..7 (cell1 = +8)
  const int cj_ = tid & 127;              // hidden index
  float cs0 = 0.f, cs1 = 0.f, hs0 = 0.f, hs1 = 0.f;

  for (int i = tid; i < 16 * HSTR; i += 1024) (&h_lds[0][0])[i] = 0.0f;
  __syncthreads();

  for (int t = 0; t < L_; ++t) {
    // stage x rows for this step (16 rows x 12 features)
    if (tid < 16 * C_IN) {
      const int r = tid / C_IN, k = tid % C_IN;
      xs[r][k] = x[((size_t)(b0 + r) * L_ + t) * C_IN + k];
    }
    __syncthreads();

    // g = x_t @ W_ihᵀ + h @ W_hhᵀ + bias  (uniform: EXEC all-ones at WMMA)
    v8f acc = {0.f, 0.f, 0.f, 0.f, 0.f, 0.f, 0.f, 0.f};
#pragma unroll
    for (int kk = 0; kk < 3; ++kk) {
      v2f a; a.x = xs[arow][4 * kk + kb]; a.y = xs[arow][4 * kk + kb + 1];
      acc = wmma4(a, bxw[kk], acc);
    }
#pragma unroll
    for (int kk = 0; kk < 32; ++kk) {
      v2f a; a.x = h_lds[arow][4 * kk + kb]; a.y = h_lds[arow][4 * kk + kb + 1];
      acc = wmma4(a, bh[kk], acc);
    }
#pragma unroll
    for (int r = 0; r < 8; ++r) g_lds[r + rhi][n] = acc[r] + biasn;
    __syncthreads();

    // gate nonlinearity + state update (cell state lives in registers)
    {
      float ig = g_lds[cb_][cj_],       fg = g_lds[cb_][cj_ + 128];
      float gg = g_lds[cb_][cj_ + 256], og = g_lds[cb_][cj_ + 384];
      cs0 = sigmoidf_(fg) * cs0 + sigmoidf_(ig) * tanhf(gg);
      const float hh0 = sigmoidf_(og) * tanhf(cs0);
      h_lds[cb_][cj_] = hh0;  hs0 += hh0;

      const int b1 = cb_ + 8;
      ig = g_lds[b1][cj_];       fg = g_lds[b1][cj_ + 128];
      gg = g_lds[b1][cj_ + 256]; og = g_lds[b1][cj_ + 384];
      cs1 = sigmoidf_(fg) * cs1 + sigmoidf_(ig) * tanhf(gg);
      const float hh1 = sigmoidf_(og) * tanhf(cs1);
      h_lds[b1][cj_] = hh1;  hs1 += hh1;
    }
    __syncthreads();
  }

  hsum_ws[(size_t)(b0 + cb_) * H_LSTM + cj_]     = hs0;
  hsum_ws[(size_t)(b0 + cb_ + 8) * H_LSTM + cj_] = hs1;
}

// ---------------------------------------------------------------------------
// Kernel 3: pooled projections + classifier head. 1 WG per batch element.
// ---------------------------------------------------------------------------
__global__ __launch_bounds__(128) void k_head(
    const float* __restrict__ ysum, const float* __restrict__ hsum,
    const float* __restrict__ out_W,
    const float* __restrict__ fc1_W, const float* __restrict__ fc1_b,
    const float* __restrict__ fc2_W, const float* __restrict__ fc2_b,
    float* __restrict__ out)
{
  const int b = blockIdx.x, m = threadIdx.x;
  __shared__ float fused[D_MODEL + H_LSTM];
  __shared__ float hfc[FC_H];
  const float invL = 1.0f / (float)L_;

  // mamba_pooled = (mean_t y) @ out_Wᵀ   (pool-then-project: linear)
  float s = 0.0f;
  const float* yb = ysum + b * D_INNER;
  const float* w0 = out_W + m * D_INNER;
  for (int d = 0; d < D_INNER; ++d) s += yb[d] * w0[d];
  fused[m]           = s * invL;
  fused[D_MODEL + m] = hsum[b * H_LSTM + m] * invL;
  __syncthreads();

  float a = fc1_b[m];
  const float* w1 = fc1_W + m * (D_MODEL + H_LSTM);
  for (int i = 0; i < D_MODEL + H_LSTM; ++i) a += fused[i] * w1[i];
  hfc[m] = fmaxf(a, 0.0f);
  __syncthreads();

  if (m < NCLS) {
    float o = fc2_b[m];
    const float* w2 = fc2_W + m * FC_H;
    for (int i = 0; i < FC_H; ++i) o += hfc[i] * w2[i];
    out[b * NCLS + m] = o;
  }
}

// ---------------------------------------------------------------------------
extern "C" void kernel_launch(void* const* d_in, const int* in_sizes, int n_in,
                              void* d_out, int out_size, void* d_ws, size_t ws_size,
                              hipStream_t stream)
{
  const float* x      = (const float*)d_in[0];
  const float* mp_W   = (const float*)d_in[1];
  const float* mp_b   = (const float*)d_in[2];
  const float* in_W   = (const float*)d_in[3];
  const float* conv_w = (const float*)d_in[4];
  const float* conv_b = (const float*)d_in[5];
  const float* xp_W   = (const float*)d_in[6];
  const float* dt_W   = (const float*)d_in[7];
  const float* dt_b   = (const float*)d_in[8];
  const float* A_log  = (const float*)d_in[9];
  const float* Dvec   = (const float*)d_in[10];
  const float* out_W  = (const float*)d_in[11];
  const float* W_ih   = (const float*)d_in[12];
  const float* W_hh   = (const float*)d_in[13];
  const float* b_ih   = (const float*)d_in[14];
  const float* b_hh   = (const float*)d_in[15];
  const float* fc1_W  = (const float*)d_in[16];
  const float* fc1_b  = (const float*)d_in[17];
  const float* fc2_W  = (const float*)d_in[18];
  const float* fc2_b  = (const float*)d_in[19];

  float* ws   = (float*)d_ws;            // ~412 KB used, all fully written
  float* Wc   = ws;                      // 512*12 = 6144
  float* bc   = ws + 6144;               // 512
  float* bsum = ws + 6656;               // 512
  float* ysum = ws + 7168;               // 256*256 = 65536
  float* hsum = ws + 72704;              // 256*128 = 32768

  k_fold <<<1, 512, 0, stream>>>(mp_W, mp_b, in_W, b_ih, b_hh, Wc, bc, bsum);
  k_mamba<<<B_, 256, 0, stream>>>(x, Wc, bc, conv_w, conv_b, xp_W, dt_W, dt_b,
                                  A_log, Dvec, ysum);
  k_lstm <<<B_ / 16, 1024, 0, stream>>>(x, W_ih, W_hh, bsum, hsum);
  k_head <<<B_, 128, 0, stream>>>(ysum, hsum, out_W, fc1_W, fc1_b, fc2_W, fc2_b,
                                  (float*)d_out);
}